// FusionFeedForward_21260088115592
// MI455X (gfx1250) — compile-verified
//
#include <hip/hip_runtime.h>
#include <hip/hip_bf16.h>
#include <math.h>

// ---------------------------------------------------------------------------
// Problem constants (from the reference): b=2, n=5760, d=512, hd=1960,
// H=60, W=108, PD=3, KH=KW=7, ST=3  ->  Hp=66, Wp=114, oh=20, ow=36, L=720,
// K=49 taps, C=40 channels, Bp=16.
// ---------------------------------------------------------------------------
#define MM      11520        // b*n rows
#define D_IN    512
#define HD      1960
#define OH      20
#define OW      36
#define LPAT    720          // OH*OW
#define CCH     40
#define KTAP    49
#define BP      16
#define HP      66
#define WP      114
#define IMG_PX  (HP*WP)      // 7524

typedef __attribute__((ext_vector_type(2))) float v2f;
typedef __attribute__((ext_vector_type(8))) float v8f;

// ---------------------------------------------------------------------------
// CDNA5 async global->LDS staging (ASYNCcnt path).  The builtin's parameters
// are GCC-vector int4 pointers: (int4 AS1* src, int4 AS3* dst, imm, imm).
// ---------------------------------------------------------------------------
#if __has_builtin(__builtin_amdgcn_global_load_async_to_lds_b128)
#define HAS_ASYNC_LDS 1
#else
#define HAS_ASYNC_LDS 0
#endif

typedef int v4i_gcc __attribute__((vector_size(16)));
typedef __attribute__((address_space(1))) v4i_gcc g_v4i;   // global int4
typedef __attribute__((address_space(3))) v4i_gcc l_v4i;   // LDS int4

__device__ __forceinline__ void copy16_g2lds(const float* gsrc, float* ldst)
{
#if HAS_ASYNC_LDS
    __builtin_amdgcn_global_load_async_to_lds_b128(
        (g_v4i*)gsrc, (l_v4i*)ldst, 0, 0);
#else
    *reinterpret_cast<float4*>(ldst) = *reinterpret_cast<const float4*>(gsrc);
#endif
}

__device__ __forceinline__ void wait_async_copies()
{
#if HAS_ASYNC_LDS
#if __has_builtin(__builtin_amdgcn_s_wait_asynccnt)
    __builtin_amdgcn_s_wait_asynccnt(0);
#else
    asm volatile("s_wait_asynccnt 0x0" ::: "memory");
#endif
#endif
}

// ---------------------------------------------------------------------------
// Generic fp32 WMMA GEMM: C[M,N] = A[M,K] @ B[K,N] + bias[N]
// block = 256 threads (8 wave32), block tile 128x64, K tile 32,
// double-buffered LDS with async global->LDS staging overlapped against the
// v_wmma_f32_16x16x4_f32 stream.  Each wave owns a 32x32 region (2x2 tiles).
// Requires M % 128 == 0, N % 4 == 0, K % 4 == 0 (true for both GEMMs here).
// ---------------------------------------------------------------------------
#define TM 128
#define TN 64
#define TK 32
#define A_STRIDE 36   // 128x32 tile padded: stride 36 -> 16 distinct banks
#define B_STRIDE 72   // 32x64 tile padded: 2*72 % 64 = 16 -> no cross-half conflict

__global__ __launch_bounds__(256)
void gemm_f32_wmma(const float* __restrict__ A, const float* __restrict__ B,
                   const float* __restrict__ bias, float* __restrict__ C,
                   int M, int N, int K)
{
    __shared__ float sA[2][TM * A_STRIDE];   // 2 x 18432 B
    __shared__ float sB[2][TK * B_STRIDE];   // 2 x  9216 B

    const int tid   = threadIdx.x;
    const int lane  = tid & 31;
    const int wave  = tid >> 5;     // 0..7
    const int waveM = wave >> 1;    // 0..3 -> 32-row slab
    const int waveN = wave & 1;     // 0..1 -> 32-col slab
    const int lm    = lane & 15;
    const int halfK = lane >> 4;    // 0/1: selects K base 0 or 2 (ISA A layout)

    const int blockM = blockIdx.y * TM;
    const int blockN = blockIdx.x * TN;

    v8f acc[2][2] = {};

    // stage one K-tile (A: 128x32 = 1024 float4, 4/thread; B: 32x64 = 512, 2/thread)
    auto stage = [&](int k0, int buf) {
        #pragma unroll
        for (int i = 0; i < 4; ++i) {
            int f4   = tid + i * 256;          // 0..1023
            int row  = f4 >> 3;                // 0..127
            int col4 = (f4 & 7) * 4;           // 0..28
            int gk   = k0 + col4;
            float* dst = &sA[buf][row * A_STRIDE + col4];
            if (gk < K)                        // K%4==0 -> whole float4 in-range
                copy16_g2lds(&A[(size_t)(blockM + row) * K + gk], dst);
            else
                *reinterpret_cast<float4*>(dst) = make_float4(0.f, 0.f, 0.f, 0.f);
        }
        #pragma unroll
        for (int i = 0; i < 2; ++i) {
            int f4   = tid + i * 256;          // 0..511
            int row  = f4 >> 4;                // 0..31
            int col4 = (f4 & 15) * 4;          // 0..60
            int gk   = k0 + row;
            int gn   = blockN + col4;
            float* dst = &sB[buf][row * B_STRIDE + col4];
            if (gk < K && gn < N)              // N%4==0 -> whole float4 in-range
                copy16_g2lds(&B[(size_t)gk * N + gn], dst);
            else
                *reinterpret_cast<float4*>(dst) = make_float4(0.f, 0.f, 0.f, 0.f);
        }
    };

    stage(0, 0);
    wait_async_copies();
    __syncthreads();

    int buf = 0;
    for (int k0 = 0; k0 < K; k0 += TK) {
        // prefetch the next tile into the other buffer while we compute
        if (k0 + TK < K)
            stage(k0 + TK, buf ^ 1);

        const float* sAb = sA[buf];
        const float* sBb = sB[buf];
        #pragma unroll
        for (int kk = 0; kk < TK; kk += 4) {
            const int kidx = kk + halfK * 2;
            v2f afrag[2], bfrag[2];
            #pragma unroll
            for (int t = 0; t < 2; ++t) {
                int arow = waveM * 32 + t * 16 + lm;
                afrag[t][0] = sAb[arow * A_STRIDE + kidx];
                afrag[t][1] = sAb[arow * A_STRIDE + kidx + 1];
                int bcol = waveN * 32 + t * 16 + lm;
                bfrag[t][0] = sBb[kidx * B_STRIDE + bcol];
                bfrag[t][1] = sBb[(kidx + 1) * B_STRIDE + bcol];
            }
            #pragma unroll
            for (int tm = 0; tm < 2; ++tm)
                #pragma unroll
                for (int tn = 0; tn < 2; ++tn)
                    acc[tm][tn] = __builtin_amdgcn_wmma_f32_16x16x4_f32(
                        false, afrag[tm], false, bfrag[tn],
                        (short)0, acc[tm][tn], false, false);
        }
        wait_async_copies();   // next tile resident in LDS
        __syncthreads();
        buf ^= 1;
    }

    // epilogue: bias add + store (C/D layout: VGPR r -> row r, +8 for lanes>=16)
    #pragma unroll
    for (int tm = 0; tm < 2; ++tm) {
        #pragma unroll
        for (int tn = 0; tn < 2; ++tn) {
            int col = blockN + waveN * 32 + tn * 16 + lm;
            if (col < N) {
                float bv = bias[col];
                #pragma unroll
                for (int r = 0; r < 8; ++r) {
                    int row = blockM + waveM * 32 + tm * 16 + r + halfK * 8;
                    C[(size_t)row * N + col] = acc[tm][tn][r] + bv;
                }
            }
        }
    }
}

// ---------------------------------------------------------------------------
// Fold: scatter-add h (Bp, L, C, Ktap) into padded images img (Bp*C, Hp*Wp).
// ---------------------------------------------------------------------------
__global__ void t2t_scatter(const float* __restrict__ h, float* __restrict__ img)
{
    size_t i = (size_t)blockIdx.x * blockDim.x + threadIdx.x;
    const size_t total = (size_t)MM * HD;
    if (i >= total) return;
    int k  = (int)(i % KTAP);
    int c  = (int)((i / KTAP) % CCH);
    int l  = (int)((i / HD) % LPAT);
    int bp = (int)(i / ((size_t)HD * LPAT));
    int ohi = l / OW, owi = l % OW;
    int ki  = k / 7,  kj  = k % 7;
    int p = (ohi * 3 + ki) * WP + (owi * 3 + kj);
    atomicAdd(&img[(size_t)(bp * CCH + c) * IMG_PX + p], h[i]);
}

// taps covering padded coordinate r along one axis with nOut output positions
__device__ __forceinline__ int tap_count(int r, int nOut)
{
    int cnt = 0;
    #pragma unroll
    for (int ki = 0; ki < 7; ++ki) {
        int q = r - ki;
        cnt += (q >= 0 && (q % 3) == 0 && (q / 3) < nOut) ? 1 : 0;
    }
    return cnt;
}

// ---------------------------------------------------------------------------
// Unfold + normalize + crop-mask + exact GELU.  Writes g in the same
// (Bp, L, C, Ktap) token layout (overwrites the h buffer — only img is read).
// ---------------------------------------------------------------------------
__global__ void t2t_gather_gelu(const float* __restrict__ img, float* __restrict__ g)
{
    size_t i = (size_t)blockIdx.x * blockDim.x + threadIdx.x;
    const size_t total = (size_t)MM * HD;
    if (i >= total) return;
    int k  = (int)(i % KTAP);
    int c  = (int)((i / KTAP) % CCH);
    int l  = (int)((i / HD) % LPAT);
    int bp = (int)(i / ((size_t)HD * LPAT));
    int ohi = l / OW, owi = l % OW;
    int ki  = k / 7,  kj  = k % 7;
    int r  = ohi * 3 + ki;   // padded row
    int cc = owi * 3 + kj;   // padded col
    float v = 0.f;
    if (r >= 3 && r < 3 + 60 && cc >= 3 && cc < 3 + 108) {
        float cnt = (float)(tap_count(r, OH) * tap_count(cc, OW));
        v = img[(size_t)(bp * CCH + c) * IMG_PX + r * WP + cc] / cnt;
    }
    // exact GELU: 0.5*x*(1+erf(x/sqrt(2)))
    g[i] = 0.5f * v * (1.f + erff(v * 0.70710678118654752f));
}

// ---------------------------------------------------------------------------
extern "C" void kernel_launch(void* const* d_in, const int* in_sizes, int n_in,
                              void* d_out, int out_size, void* d_ws, size_t ws_size,
                              hipStream_t stream)
{
    (void)in_sizes; (void)n_in; (void)out_size; (void)ws_size;

    const float* x  = (const float*)d_in[0];   // (2,5760,512)
    const float* W1 = (const float*)d_in[1];   // (512,1960)
    const float* b1 = (const float*)d_in[2];   // (1960,)
    const float* W2 = (const float*)d_in[3];   // (1960,512)
    const float* b2 = (const float*)d_in[4];   // (512,)
    // d_in[5]=out_h, d_in[6]=out_w : fixed 60/108 per setup_inputs

    float* h_g = (float*)d_ws;                                   // 11520*1960 floats (h, then g)
    float* img = (float*)((char*)d_ws + (size_t)MM * HD * sizeof(float)); // 640*7524 floats
    float* out = (float*)d_out;                                  // 11520*512 floats

    const dim3 blk(256);

    // GEMM1: h = x @ W1 + b1   (M=11520, N=1960, K=512)
    dim3 g1((HD + TN - 1) / TN, MM / TM);
    gemm_f32_wmma<<<g1, blk, 0, stream>>>(x, W1, b1, h_g, MM, HD, D_IN);

    // fold: zero image, scatter-add
    (void)hipMemsetAsync(img, 0, (size_t)BP * CCH * IMG_PX * sizeof(float), stream);
    size_t total = (size_t)MM * HD;
    unsigned nb = (unsigned)((total + 255) / 256);
    t2t_scatter<<<nb, blk, 0, stream>>>(h_g, img);

    // unfold + normalize + gelu (g overwrites h)
    t2t_gather_gelu<<<nb, blk, 0, stream>>>(img, h_g);

    // GEMM2: out = g @ W2 + b2   (M=11520, N=512, K=1960)
    dim3 g2(D_IN / TN, MM / TM);
    gemm_f32_wmma<<<g2, blk, 0, stream>>>(h_g, W2, b2, out, MM, D_IN, HD);
}